// Net_37692632990403
// MI455X (gfx1250) — compile-verified
//
#include <hip/hip_runtime.h>

// ---------------------------------------------------------------------------
// SNN forward (snntorch Synaptic x2), restructured for CDNA5 / gfx1250:
//   1) GEMM1: CUR1[16000,512] = X[16000,784] @ W1^T + b1   (fp32 WMMA 16x16x4)
//   2) SCAN1: per-(b,h) recurrence over t -> spk1_rec, mem1_rec
//   3) GEMM2: CUR2[16000,10]  = SPK1[16000,512] @ W2^T + b2 (fp32 WMMA, N padded to 16)
//   4) SCAN2: per-(b,o) recurrence over t -> spk2_rec, mem2_rec
// CUR1/CUR2 are staged in the mem1_rec/mem2_rec output slots (identical shape)
// and overwritten in place by the scans (read-before-write per element).
// ---------------------------------------------------------------------------

typedef float v2f __attribute__((ext_vector_type(2)));
typedef float v8f __attribute__((ext_vector_type(8)));

#define T_STEPS 1000
#define BATCH   16
#define IN_DIM  784
#define HID     512
#define OUT_DIM 10

// ---------------- GEMM1: one wave per 16x16 tile, K stepped by 4 -----------
// A: X rows (M = t*16+b), B: W1 rows (N), both K-contiguous -> float2 loads
// match the 32-bit WMMA A/B lane layouts (lanes 0-15: K={k,k+1};
// lanes 16-31: K={k+2,k+3}).
__global__ void snn_gemm1(const float* __restrict__ X,   // [16000, 784]
                          const float* __restrict__ W1,  // [512, 784]
                          const float* __restrict__ b1,  // [512]
                          float* __restrict__ cur1)      // [16000, 512]
{
    const int wave   = blockIdx.x * (blockDim.x >> 5) + (threadIdx.x >> 5);
    const int lane   = threadIdx.x & 31;
    const int tile_n = wave & 31;   // 32 N-tiles (512/16)
    const int tile_m = wave >> 5;   // 1000 M-tiles (16000/16)
    const int half   = lane >> 4;   // 0: lanes 0-15, 1: lanes 16-31
    const int lmn    = lane & 15;   // M index for A, N index for B/C/D

    const float* arow = X  + (size_t)(tile_m * 16 + lmn) * IN_DIM;
    const float* brow = W1 + (size_t)(tile_n * 16 + lmn) * IN_DIM;

    v8f c = {0.f, 0.f, 0.f, 0.f, 0.f, 0.f, 0.f, 0.f};

#pragma unroll 4
    for (int k = 0; k < IN_DIM; k += 4) {
        const int kk = k + 2 * half;
        v2f a = *(const v2f*)(arow + kk);   // A[m, kk], A[m, kk+1]
        v2f b = *(const v2f*)(brow + kk);   // B[kk, n], B[kk+1, n] = W1[n, kk..]
        c = __builtin_amdgcn_wmma_f32_16x16x4_f32(
                /*neg_a=*/false, a, /*neg_b=*/false, b,
                /*c_mod=*/(short)0, c, /*reuse_a=*/false, /*reuse_b=*/false);
    }

    const float bias = b1[tile_n * 16 + lmn];
    // C/D layout: VGPR r -> row (r + 8*half), col lmn
    float* out = cur1 + (size_t)(tile_m * 16 + 8 * half) * HID + tile_n * 16 + lmn;
#pragma unroll
    for (int r = 0; r < 8; ++r)
        out[(size_t)r * HID] = c[r] + bias;
}

// ---------------- SCAN1: 8192 independent recurrences, coalesced ----------
// curmem aliases: reads CUR1[t] then overwrites same element with mem1_rec[t].
__global__ void snn_scan1(float* curmem,                  // [1000,16,512] in/out
                          float* __restrict__ spk1)       // [1000,16,512]
{
    const int idx = blockIdx.x * blockDim.x + threadIdx.x;  // b*512 + h
    float syn = 0.f, mem = 0.f;
    for (int t = 0; t < T_STEPS; ++t) {
        const size_t off = (size_t)t * (BATCH * HID) + idx;
        const float cur = curmem[off];
        const float rst = (mem > 1.0f) ? 1.0f : 0.0f;  // from mem at step entry
        syn = 0.9f * syn + cur;                        // ALPHA
        mem = 0.8f * mem + syn - rst;                  // BETA, THR=1
        curmem[off] = mem;                             // mem1_rec
        spk1[off]   = (mem - 1.0f > 0.0f) ? 1.0f : 0.0f;
    }
}

// ---------------- GEMM2: one wave per M-tile, N padded 10 -> 16 ------------
__global__ void snn_gemm2(const float* __restrict__ SPK1, // [16000, 512]
                          const float* __restrict__ W2,   // [10, 512]
                          const float* __restrict__ b2,   // [10]
                          float* __restrict__ cur2)       // [16000, 10]
{
    const int tile_m = blockIdx.x * (blockDim.x >> 5) + (threadIdx.x >> 5); // 0..999
    const int lane   = threadIdx.x & 31;
    const int half   = lane >> 4;
    const int lmn    = lane & 15;

    const float* arow = SPK1 + (size_t)(tile_m * 16 + lmn) * HID;
    // Arithmetic masking for padded N (keeps EXEC uniform for WMMA):
    const int   nsafe  = (lmn < OUT_DIM) ? lmn : 0;
    const float bscale = (lmn < OUT_DIM) ? 1.0f : 0.0f;
    const float* brow  = W2 + (size_t)nsafe * HID;

    v8f c = {0.f, 0.f, 0.f, 0.f, 0.f, 0.f, 0.f, 0.f};

#pragma unroll 4
    for (int k = 0; k < HID; k += 4) {
        const int kk = k + 2 * half;
        v2f a = *(const v2f*)(arow + kk);
        v2f b = *(const v2f*)(brow + kk);
        b.x *= bscale;
        b.y *= bscale;
        c = __builtin_amdgcn_wmma_f32_16x16x4_f32(
                false, a, false, b, (short)0, c, false, false);
    }

    if (lmn < OUT_DIM) {
        const float bias = b2[lmn];
        float* out = cur2 + (size_t)(tile_m * 16 + 8 * half) * OUT_DIM + lmn;
#pragma unroll
        for (int r = 0; r < 8; ++r)
            out[(size_t)r * OUT_DIM] = c[r] + bias;
    }
}

// ---------------- SCAN2: 160 independent recurrences ----------------------
__global__ void snn_scan2(float* curmem,                  // [1000,16,10] in/out
                          float* __restrict__ spk2)       // [1000,16,10]
{
    const int idx = threadIdx.x;                 // b*10 + o, 0..159
    if (idx >= BATCH * OUT_DIM) return;
    float syn = 0.f, mem = 0.f;
    for (int t = 0; t < T_STEPS; ++t) {
        const size_t off = (size_t)t * (BATCH * OUT_DIM) + idx;
        const float cur = curmem[off];
        const float rst = (mem > 1.0f) ? 1.0f : 0.0f;
        syn = 0.9f * syn + cur;
        mem = 0.8f * mem + syn - rst;
        curmem[off] = mem;
        spk2[off]   = (mem - 1.0f > 0.0f) ? 1.0f : 0.0f;
    }
}

extern "C" void kernel_launch(void* const* d_in, const int* in_sizes, int n_in,
                              void* d_out, int out_size, void* d_ws, size_t ws_size,
                              hipStream_t stream) {
    (void)in_sizes; (void)n_in; (void)out_size; (void)d_ws; (void)ws_size;

    const float* x  = (const float*)d_in[0];   // [1000,16,784]
    const float* W1 = (const float*)d_in[1];   // [512,784]
    const float* b1 = (const float*)d_in[2];   // [512]
    const float* W2 = (const float*)d_in[3];   // [10,512]
    const float* b2 = (const float*)d_in[4];   // [10]

    float* out = (float*)d_out;
    const size_t n1 = (size_t)T_STEPS * BATCH * HID;      // 8,192,000
    const size_t n2 = (size_t)T_STEPS * BATCH * OUT_DIM;  // 160,000
    float* spk1 = out;                 // [1000,16,512]
    float* mem1 = out + n1;            // staging for CUR1, then mem1_rec
    float* spk2 = out + 2 * n1;        // [1000,16,10]
    float* mem2 = out + 2 * n1 + n2;   // staging for CUR2, then mem2_rec

    // 1) CUR1 = X @ W1^T + b1  -> staged in mem1_rec slot
    //    32000 tiles (1000 M x 32 N), 4 waves/block
    snn_gemm1<<<dim3(32000 / 4), dim3(128), 0, stream>>>(x, W1, b1, mem1);

    // 2) layer-1 recurrence (8192 lanes)
    snn_scan1<<<dim3((BATCH * HID) / 256), dim3(256), 0, stream>>>(mem1, spk1);

    // 3) CUR2 = SPK1 @ W2^T + b2 -> staged in mem2_rec slot (1000 M-tiles)
    snn_gemm2<<<dim3(1000 / 4), dim3(128), 0, stream>>>(spk1, W2, b2, mem2);

    // 4) layer-2 recurrence (160 lanes)
    snn_scan2<<<dim3(1), dim3(160), 0, stream>>>(mem2, spk2);
}